// TransferEntropy_23587960390352
// MI455X (gfx1250) — compile-verified
//
#include <hip/hip_runtime.h>
#include <hip/hip_bf16.h>

// Problem constants (from reference): B=16, F=512, D=1, T=512, LAG=1, MORDER=1
#define B_   16
#define F_   512
#define T_   512
#define NV_  511          // T - MORDER*LAG
#define BF_  (B_ * F_)    // 8192 rows
#define EPS_ 1e-6f

static const size_t NPLANE = (size_t)B_ * F_ * T_;   // 4,194,304 floats per plane

typedef __attribute__((ext_vector_type(2))) float v2f;
typedef __attribute__((ext_vector_type(8))) float v8f;

// ---------------- block reductions (blockDim.x == 512) ----------------
__device__ __forceinline__ float blk_sum(float v, float* s) {
    int t = threadIdx.x;
    __syncthreads();            // protect previous consumers of s
    s[t] = v;
    __syncthreads();
    #pragma unroll
    for (int off = 256; off > 0; off >>= 1) {
        if (t < off) s[t] += s[t + off];
        __syncthreads();
    }
    return s[0];
}

__device__ __forceinline__ float blk_max(float v, float* s) {
    int t = threadIdx.x;
    __syncthreads();
    s[t] = v;
    __syncthreads();
    #pragma unroll
    for (int off = 256; off > 0; off >>= 1) {
        if (t < off) s[t] = fmaxf(s[t], s[t + off]);
        __syncthreads();
    }
    return s[0];
}

// ---------------- f32 WMMA wrapper (V_WMMA_F32_16X16X4_F32) ----------------
__device__ __forceinline__ v8f wmma_f32(v2f a, v2f b, v8f c) {
    // 8 args: (neg_a, A, neg_b, B, c_mod, C, reuse_a, reuse_b)
    return __builtin_amdgcn_wmma_f32_16x16x4_f32(
        false, a, false, b, (short)0, c, false, false);
}

// ============================================================================
// Kernel 1: preprocess queries & keys.
//   tsd -> standardize over T -> embedded rows (mean-centered over nv=511)
//   zq0[t] = qs[t]   (t<511), zq1[t] = qs[t+1], zk[t] = ks[t]; index 511 = 0 pad
//   diag: a=<zq0,zq0>/510, b=<zq0,zq1>/510, c=<zq1,zq1>/510, d=<zk,zk>/510
// one block (512 threads) per (b,f)
// ============================================================================
__global__ void __launch_bounds__(512)
prep_kernel(const float* __restrict__ q, const float* __restrict__ kk,
            float* __restrict__ zq0, float* __restrict__ zq1,
            float* __restrict__ zk, float* __restrict__ diag) {
    __shared__ float s_d[512];
    __shared__ float s_r[512];
    const int bf = blockIdx.x;
    const int t  = threadIdx.x;
    const size_t rowoff = (size_t)bf * T_;

    // ---------------- queries ----------------
    {
        float x = q[rowoff + t];
        s_d[t] = x;
        __syncthreads();
        float d = (t == 0) ? (s_d[1] - s_d[0]) : (s_d[t] - s_d[t - 1]);
        float m = blk_sum(d, s_r) * (1.0f / T_);
        float c = d - m;
        float ss = blk_sum(c * c, s_r) * (1.0f / T_);
        float v = c * rsqrtf(ss);                 // standardized tsd series
        __syncthreads();
        s_d[t] = v;
        __syncthreads();
        float r0 = (t < NV_) ? v : 0.0f;          // qs[0..510]
        float r1 = (t < NV_) ? s_d[t + 1] : 0.0f; // qs[1..511]
        float m0 = blk_sum(r0, s_r) * (1.0f / NV_);
        float m1 = blk_sum(r1, s_r) * (1.0f / NV_);
        float z0 = (t < NV_) ? (r0 - m0) : 0.0f;
        float z1 = (t < NV_) ? (r1 - m1) : 0.0f;
        zq0[rowoff + t] = z0;
        zq1[rowoff + t] = z1;
        float aa = blk_sum(z0 * z0, s_r);
        float bb = blk_sum(z0 * z1, s_r);
        float cc = blk_sum(z1 * z1, s_r);
        if (t == 0) {
            diag[0 * BF_ + bf] = aa * (1.0f / 510.0f);
            diag[1 * BF_ + bf] = bb * (1.0f / 510.0f);
            diag[2 * BF_ + bf] = cc * (1.0f / 510.0f);
        }
    }
    // ---------------- keys ----------------
    {
        float x = kk[rowoff + t];
        __syncthreads();
        s_d[t] = x;
        __syncthreads();
        float d = (t == 0) ? (s_d[1] - s_d[0]) : (s_d[t] - s_d[t - 1]);
        float m = blk_sum(d, s_r) * (1.0f / T_);
        float c = d - m;
        float ss = blk_sum(c * c, s_r) * (1.0f / T_);
        float v = c * rsqrtf(ss);
        float r0 = (t < NV_) ? v : 0.0f;
        float m0 = blk_sum(r0, s_r) * (1.0f / NV_);
        float z = (t < NV_) ? (r0 - m0) : 0.0f;
        zk[rowoff + t] = z;
        float dd = blk_sum(z * z, s_r);
        if (t == 0) diag[3 * BF_ + bf] = dd * (1.0f / 510.0f);
    }
}

// ============================================================================
// Kernel 2: E[b,f,g] = <zq0_f, zk_g>/510 ; H[b,f,g] = <zq1_f, zk_g>/510
// One wave per 16x16 (f,g) tile per batch, f32 WMMA, K stepped by 4.
// A frag (16x4): lane<16 -> row M=lane, K=0..1 ; lane>=16 -> row M=lane-16, K=2..3
// B frag (4x16): lane<16 -> col N=lane, K=0..1 ; lane>=16 -> col N=lane-16, K=2..3
//   B[k][n] = zk[g0+n][k]  (Gram: B = Zk^T), so both frags are float2 row loads.
// C (16x16): vgpr r: row = r + 8*(lane>=16), col = lane&15.
// ============================================================================
__global__ void __launch_bounds__(128)
gemm_eh_kernel(const float* __restrict__ zq0, const float* __restrict__ zq1,
               const float* __restrict__ zk,
               float* __restrict__ E, float* __restrict__ H) {
    const int gwave = (blockIdx.x * blockDim.x + threadIdx.x) >> 5;
    const int lane  = threadIdx.x & 31;
    const int b   = gwave >> 10;          // / (32*32)
    const int rem = gwave & 1023;
    const int f0  = (rem >> 5) << 4;
    const int g0  = (rem & 31) << 4;

    const int mrow = lane & 15;
    const int ksub = (lane >> 4) << 1;    // 0 or 2

    const float* A0 = zq0 + ((size_t)b * F_ + f0 + mrow) * T_ + ksub;
    const float* A1 = zq1 + ((size_t)b * F_ + f0 + mrow) * T_ + ksub;
    const float* Bk = zk  + ((size_t)b * F_ + g0 + mrow) * T_ + ksub;

    v8f c0 = {}; v8f c1 = {};
    #pragma unroll 4
    for (int k = 0; k < T_; k += 4) {
        v2f a0 = *(const v2f*)(A0 + k);
        v2f a1 = *(const v2f*)(A1 + k);
        v2f bb = *(const v2f*)(Bk + k);
        c0 = wmma_f32(a0, bb, c0);
        c1 = wmma_f32(a1, bb, c1);
    }

    const int col   = lane & 15;
    const int rbase = (lane >> 4) << 3;
    #pragma unroll
    for (int r = 0; r < 8; ++r) {
        size_t idx = ((size_t)b * F_ + f0 + rbase + r) * (size_t)F_ + g0 + col;
        E[idx] = c0[r] * (1.0f / 510.0f);
        H[idx] = c1[r] * (1.0f / 510.0f);
    }
}

// ============================================================================
// Kernel 3: pte via closed-form 2x2 / 3x3 dets, then row softmax.
// attn written in-place into E. One 512-thread block per (b,f) row.
// ============================================================================
__global__ void __launch_bounds__(512)
pte_softmax_kernel(float* __restrict__ E, const float* __restrict__ H,
                   const float* __restrict__ diag) {
    __shared__ float s_r[512];
    const int b = blockIdx.x / F_;
    const int f = blockIdx.x % F_;
    const int g = threadIdx.x;

    const float a  = diag[0 * BF_ + b * F_ + f];
    const float bb = diag[1 * BF_ + b * F_ + f];
    const float cc = diag[2 * BF_ + b * F_ + f];
    const float dd = diag[3 * BF_ + b * F_ + g];

    const size_t idx = ((size_t)b * F_ + f) * (size_t)F_ + g;
    const float e = E[idx];
    const float h = H[idx];

    // H_XtYt = |det [[a, e],[e, dd]]| ; H_YYt = |det [[a, bb],[bb, cc]]|
    // H_YYtXt = |det [[a, bb, e],[bb, cc, h],[e, h, dd]]| ; H_Yt = a (1x1, no abs)
    const float H_XtYt = fabsf(a * dd - e * e);
    const float H_YYt  = fabsf(a * cc - bb * bb);
    const float det3 = a * (cc * dd - h * h)
                     - bb * (bb * dd - h * e)
                     + e  * (bb * h  - cc * e);
    const float H_YYtXt = fabsf(det3);
    const float pte = 0.5f * (logf(H_XtYt / (H_YYtXt + EPS_) + 1.0f)
                            - logf(a      / (H_YYt   + EPS_) + 1.0f));

    const float mx = blk_max(pte, s_r);
    const float ex = expf(pte - mx);
    const float sm = blk_sum(ex, s_r);
    E[idx] = ex / sm;        // attn in place
}

// ============================================================================
// Kernel 4: out[b,f,h] = (attn @ V)[b,f,h]*kg + V[b,f,h]*(1-kg), kg=sigmoid(k)
// Same WMMA tiling; K dim = g (512). B frag = V[b][k][h] (row-major, K strided).
// ============================================================================
__global__ void __launch_bounds__(128)
out_gemm_kernel(const float* __restrict__ attn, const float* __restrict__ V,
                const float* __restrict__ kptr, float* __restrict__ out) {
    const int gwave = (blockIdx.x * blockDim.x + threadIdx.x) >> 5;
    const int lane  = threadIdx.x & 31;
    const int b   = gwave >> 10;
    const int rem = gwave & 1023;
    const int f0  = (rem >> 5) << 4;
    const int h0  = (rem & 31) << 4;

    const int mrow = lane & 15;
    const int ksub = (lane >> 4) << 1;

    const float* Arow = attn + ((size_t)b * F_ + f0 + mrow) * (size_t)F_ + ksub;
    const float* Vb   = V + (size_t)b * F_ * T_;
    const int    ncol = lane & 15;

    v8f c = {};
    #pragma unroll 4
    for (int k = 0; k < F_; k += 4) {
        v2f a = *(const v2f*)(Arow + k);
        v2f bv;
        bv.x = Vb[(size_t)(k + ksub + 0) * T_ + h0 + ncol];
        bv.y = Vb[(size_t)(k + ksub + 1) * T_ + h0 + ncol];
        c = wmma_f32(a, bv, c);
    }

    const float kv = kptr[0];
    const float kg = 1.0f / (1.0f + expf(-kv));

    const int rbase = (lane >> 4) << 3;
    #pragma unroll
    for (int r = 0; r < 8; ++r) {
        size_t idx = ((size_t)b * F_ + f0 + rbase + r) * (size_t)T_ + h0 + ncol;
        out[idx] = c[r] * kg + V[idx] * (1.0f - kg);
    }
}

// ============================================================================
extern "C" void kernel_launch(void* const* d_in, const int* in_sizes, int n_in,
                              void* d_out, int out_size, void* d_ws, size_t ws_size,
                              hipStream_t stream) {
    const float* q  = (const float*)d_in[0];   // (16,512,1,512) f32
    const float* kk = (const float*)d_in[1];   // (16,512,1,512) f32
    const float* vv = (const float*)d_in[2];   // (16,512,1,512) f32
    const float* ks = (const float*)d_in[3];   // scalar k
    float* out = (float*)d_out;

    // workspace layout (floats):
    //   zq0 | zq1 | zk : NPLANE each (rows padded to 512, element 511 = 0)
    //   diag           : 4 * BF_
    //   E | H          : NPLANE each (E reused as attn after kernel 3)
    float* ws   = (float*)d_ws;
    float* zq0  = ws;
    float* zq1  = zq0 + NPLANE;
    float* zk   = zq1 + NPLANE;
    float* diag = zk + NPLANE;
    float* E    = diag + 4 * BF_;
    float* H    = E + NPLANE;
    // total: 5*NPLANE + 4*BF_ floats ~= 84 MB

    // 1) preprocess: one block per (b,f)
    prep_kernel<<<BF_, 512, 0, stream>>>(q, kk, zq0, zq1, zk, diag);

    // 2) E/H Gram GEMMs: 16 batches * 32*32 tiles, 4 waves/block
    const int tiles = B_ * 32 * 32;            // 16384 waves
    gemm_eh_kernel<<<tiles / 4, 128, 0, stream>>>(zq0, zq1, zk, E, H);

    // 3) pte + softmax (attn in-place into E)
    pte_softmax_kernel<<<BF_, 512, 0, stream>>>(E, H, diag);

    // 4) attn @ V, blended with V
    out_gemm_kernel<<<tiles / 4, 128, 0, stream>>>(E, vv, ks, out);
}